// KalmanNetNN_1941325217939
// MI455X (gfx1250) — compile-verified
//
#include <hip/hip_runtime.h>
#include <hip/hip_bf16.h>
#include <math.h>

// ---------------- problem constants ----------------
#define MM   24
#define NN   12
#define H1   2880
#define HIDN 7200
#define H2   1152
#define DOUT 288

// ---------------- workspace layout (floats) ----------------
#define OFF_KGIN   0        // 36
#define OFF_PRIOR  64       // 24
#define OFF_DY     96       // 12
#define OFF_L1     128      // 2880
#define OFF_GI     3072     // 21600  (Wih@l1 + bih)
#define OFF_GH     24704    // 21600  (Whh@hn + bhh)
#define OFF_H      46336    // 7200   (h_new)
#define OFF_L2     53568    // 1152
#define OFF_KG     54784    // 288
// total ~55072 floats = ~215 KB of d_ws

typedef float v2f __attribute__((ext_vector_type(2)));
typedef float v4f __attribute__((ext_vector_type(4)));
typedef float v8f __attribute__((ext_vector_type(8)));

#define GEMV_WAVES 8

static __device__ __forceinline__ v8f wmma_f32x4(v2f a, v2f b, v8f c) {
    return __builtin_amdgcn_wmma_f32_16x16x4_f32(
        /*neg_a=*/false, a, /*neg_b=*/false, b,
        /*c_mod=*/(short)0, c, /*reuse_a=*/false, /*reuse_b=*/false);
}

// ============================================================
// Phase 1: priors, innovation, l2-normalized feature vector
// ============================================================
__global__ void knet_phase1(const float* __restrict__ yt,
                            const float* __restrict__ F,
                            const float* __restrict__ Hm,
                            const float* __restrict__ post,
                            const float* __restrict__ pold,
                            float* __restrict__ ws)
{
    __shared__ float s_prior[MM];
    __shared__ float s_m1y[NN];
    __shared__ float s_inv[2];
    const int t = threadIdx.x;

    if (t < MM) {
        float a = 0.f;
        for (int k = 0; k < MM; ++k) a += F[t * MM + k] * post[k];
        s_prior[t] = a;
        ws[OFF_PRIOR + t] = a;
    }
    __syncthreads();
    if (t < NN) {
        float a = 0.f;
        for (int k = 0; k < MM; ++k) a += Hm[t * MM + k] * s_prior[k];
        s_m1y[t] = a;
        ws[OFF_DY + t] = yt[t] - a;
    }
    __syncthreads();
    if (t == 0) {
        float sx = 0.f;
        for (int i = 0; i < MM; ++i) { float d = post[i] - pold[i]; sx += d * d; }
        s_inv[0] = 1.f / fmaxf(sqrtf(sx), 1e-12f);
        float sy = 0.f;
        for (int i = 0; i < NN; ++i) { float d = yt[i] - s_m1y[i]; sy += d * d; }
        s_inv[1] = 1.f / fmaxf(sqrtf(sy), 1e-12f);
    }
    __syncthreads();
    if (t < NN) ws[OFF_KGIN + t] = (yt[t] - s_m1y[t]) * s_inv[1];
    if (t < MM) ws[OFF_KGIN + NN + t] = (post[t] - pold[t]) * s_inv[0];
}

// ============================================================
// fp32 GEMV via V_WMMA_F32_16X16X4_F32.
//   out[m0..m0+15] = act(W[m0..m0+15, :] @ x + b)
// One block (256 thr = 8 waves) per 16-row tile; waves split K,
// partial 16x16 accumulators reduced through LDS.
//
// Inner loop uses a K-permutation so each lane does full 16B loads:
// per k-step of 8, lane-half h loads float4 at (row*K + k + 4h).
// WMMA#0 takes elements {0,1} (K = k,k+1 | k+4,k+5) and WMMA#1 takes
// elements {2,3} (K = k+2,k+3 | k+6,k+7); B (= x, broadcast over all
// 16 columns) uses the identical mapping, so the dot product is exact.
// Every column of D equals the GEMV result; we read column 0.
// ============================================================
template <bool NT, int ACT>
__global__ __launch_bounds__(256) void knet_gemv_wmma(
    const float* __restrict__ W, const float* __restrict__ x,
    const float* __restrict__ bias, float* __restrict__ out, int K)
{
    __shared__ float red[GEMV_WAVES * 32 * 8];   // 8 KB

    const int tid  = threadIdx.x;
    const int wave = tid >> 5;
    const int lane = tid & 31;
    const int half = lane >> 4;
    const int l15  = lane & 15;     // row within tile (A) / column (B,D)
    const int m0   = blockIdx.x * 16;

    // per-wave K chunk, multiple of 4 (all K here are multiples of 4)
    const int kc = ((K / 4 + GEMV_WAVES - 1) / GEMV_WAVES) * 4;
    const int k0 = wave * kc;
    const int k1 = (k0 + kc < K) ? (k0 + kc) : K;

    const size_t rowbase = (size_t)(m0 + l15) * (size_t)K;
    const float* __restrict__ Wp4 = W + rowbase + 4 * half;  // 16B-aligned at k%4==0
    const float* __restrict__ Wp2 = W + rowbase + 2 * half;  // 8B-aligned tail path
    const float* __restrict__ x4  = x + 4 * half;
    const float* __restrict__ x2  = x + 2 * half;

    v8f acc = {0.f, 0.f, 0.f, 0.f, 0.f, 0.f, 0.f, 0.f};

    int k = k0;
    // ---- main: 16 K per iter, 2x b128 W loads, 4 WMMAs ----
    for (; k + 16 <= k1; k += 16) {
        v4f a0, a1;
        if (NT) {
            a0 = __builtin_nontemporal_load((const v4f*)(Wp4 + k));
            a1 = __builtin_nontemporal_load((const v4f*)(Wp4 + k + 8));
        } else {
            a0 = *(const v4f*)(Wp4 + k);
            a1 = *(const v4f*)(Wp4 + k + 8);
        }
        const v4f b0 = *(const v4f*)(x4 + k);
        const v4f b1 = *(const v4f*)(x4 + k + 8);
        acc = wmma_f32x4((v2f){a0.x, a0.y}, (v2f){b0.x, b0.y}, acc);
        acc = wmma_f32x4((v2f){a0.z, a0.w}, (v2f){b0.z, b0.w}, acc);
        acc = wmma_f32x4((v2f){a1.x, a1.y}, (v2f){b1.x, b1.y}, acc);
        acc = wmma_f32x4((v2f){a1.z, a1.w}, (v2f){b1.z, b1.w}, acc);
    }
    // ---- tail: 8 K per iter, 1x b128 W load, 2 WMMAs ----
    for (; k + 8 <= k1; k += 8) {
        v4f a0;
        if (NT) a0 = __builtin_nontemporal_load((const v4f*)(Wp4 + k));
        else    a0 = *(const v4f*)(Wp4 + k);
        const v4f b0 = *(const v4f*)(x4 + k);
        acc = wmma_f32x4((v2f){a0.x, a0.y}, (v2f){b0.x, b0.y}, acc);
        acc = wmma_f32x4((v2f){a0.z, a0.w}, (v2f){b0.z, b0.w}, acc);
    }
    // ---- tail: 4 K, classic layout (half -> K offset 2h) ----
    for (; k < k1; k += 4) {
        v2f a;
        if (NT) a = __builtin_nontemporal_load((const v2f*)(Wp2 + k));
        else    a = *(const v2f*)(Wp2 + k);
        const v2f b = *(const v2f*)(x2 + k);
        acc = wmma_f32x4(a, b, acc);
    }

    // cross-wave reduction of the 16x16 partial tiles
    *(v8f*)&red[(wave * 32 + lane) * 8] = acc;
    __syncthreads();

    const int rlane = tid & 31;     // lane within the D tile
    const int v     = tid >> 5;     // accumulator VGPR index 0..7
    float s = 0.f;
#pragma unroll
    for (int w = 0; w < GEMV_WAVES; ++w) s += red[(w * 32 + rlane) * 8 + v];

    // D layout: VGPR v, lanes 0-15 -> M=v ; lanes 16-31 -> M=v+8. Column = rlane&15.
    if ((rlane & 15) == 0) {
        const int row = v + 8 * (rlane >> 4);
        float r = s + bias[m0 + row];
        if (ACT == 1) r = fmaxf(r, 0.f);
        out[m0 + row] = r;
    }
}

// ============================================================
// GRU pointwise cell:  gates in ws (gi, gh), h from input hn
// ============================================================
__global__ void knet_gru(const float* __restrict__ gi,
                         const float* __restrict__ gh,
                         const float* __restrict__ h,
                         float* __restrict__ hnew)
{
    const int i = blockIdx.x * blockDim.x + threadIdx.x;
    if (i >= HIDN) return;
    const float r = 1.f / (1.f + expf(-(gi[i]            + gh[i])));
    const float z = 1.f / (1.f + expf(-(gi[HIDN + i]     + gh[HIDN + i])));
    const float n = tanhf(gi[2 * HIDN + i] + r * gh[2 * HIDN + i]);
    hnew[i] = (1.f - z) * n + z * h[i];
}

// ============================================================
// Final: m1x_post = prior + kg(24x12) @ dy(12)
// ============================================================
__global__ void knet_final(const float* __restrict__ kg,
                           const float* __restrict__ ws,
                           float* __restrict__ out)
{
    const int t = threadIdx.x;
    if (t < MM) {
        float a = ws[OFF_PRIOR + t];
        for (int n = 0; n < NN; ++n) a += kg[t * NN + n] * ws[OFF_DY + n];
        out[t] = a;
    }
}

extern "C" void kernel_launch(void* const* d_in, const int* in_sizes, int n_in,
                              void* d_out, int out_size, void* d_ws, size_t ws_size,
                              hipStream_t stream) {
    const float* yt   = (const float*)d_in[0];
    const float* F    = (const float*)d_in[1];
    const float* Hm   = (const float*)d_in[2];
    const float* post = (const float*)d_in[3];
    const float* pold = (const float*)d_in[4];
    const float* hn   = (const float*)d_in[5];
    const float* W1   = (const float*)d_in[6];
    const float* b1   = (const float*)d_in[7];
    const float* Wih  = (const float*)d_in[8];
    const float* Whh  = (const float*)d_in[9];
    const float* bih  = (const float*)d_in[10];
    const float* bhh  = (const float*)d_in[11];
    const float* W2   = (const float*)d_in[12];
    const float* b2   = (const float*)d_in[13];
    const float* W3   = (const float*)d_in[14];
    const float* b3   = (const float*)d_in[15];
    float* ws  = (float*)d_ws;
    float* out = (float*)d_out;

    // 1) priors + normalized KGain input (tiny, 1 block)
    knet_phase1<<<1, 64, 0, stream>>>(yt, F, Hm, post, pold, ws);

    // 2) l1 = relu(W1 @ kg_in + b1)                 2880 x 36
    knet_gemv_wmma<false, 1><<<H1 / 16, 256, 0, stream>>>(
        W1, ws + OFF_KGIN, b1, ws + OFF_L1, MM + NN);

    // 3a) gi = Wih @ l1 + bih                       21600 x 2880  (NT stream)
    knet_gemv_wmma<true, 0><<<(3 * HIDN) / 16, 256, 0, stream>>>(
        Wih, ws + OFF_L1, bih, ws + OFF_GI, H1);

    // 3b) gh = Whh @ hn + bhh                       21600 x 7200  (NT stream)
    knet_gemv_wmma<true, 0><<<(3 * HIDN) / 16, 256, 0, stream>>>(
        Whh, hn, bhh, ws + OFF_GH, HIDN);

    // 3c) GRU pointwise -> h_new
    knet_gru<<<(HIDN + 255) / 256, 256, 0, stream>>>(
        ws + OFF_GI, ws + OFF_GH, hn, ws + OFF_H);

    // 4) l2 = relu(W2 @ h_new + b2)                 1152 x 7200   (NT stream)
    knet_gemv_wmma<true, 1><<<H2 / 16, 256, 0, stream>>>(
        W2, ws + OFF_H, b2, ws + OFF_L2, HIDN);

    // 5) kg = W3 @ l2 + b3                          288 x 1152
    knet_gemv_wmma<false, 0><<<DOUT / 16, 256, 0, stream>>>(
        W3, ws + OFF_L2, b3, ws + OFF_KG, H2);

    // 6) m1x_post = prior + kg @ dy (tiny)
    knet_final<<<1, 32, 0, stream>>>(ws + OFF_KG, ws, out);
}